// Gate_36412732735547
// MI455X (gfx1250) — compile-verified
//
#include <hip/hip_runtime.h>

// ---------------------------------------------------------------------------
// MI455X (gfx1250) implementation of conv-gate + top-k mask + upsampled multiply.
//
// Pipeline (all memory-bound; ~610 MB HBM traffic => ~26us floor @ 23.3 TB/s):
//   K1 gate_conv_wmma : g[64][128][128] = conv4x4s4(in, wk)  via V_WMMA_F32_16X16X4_F32,
//                       inputs staged to LDS with GLOBAL_LOAD_ASYNC_TO_LDS_B128 (ASYNCcnt)
//   K2 gate_select    : per-batch 4096th-largest threshold key (256-bin MSD radix select)
//   K3 gate_apply     : out = in * (key(g) >= thr[b] ? g : 0), b128 streaming, NT stores
// ---------------------------------------------------------------------------

typedef __attribute__((ext_vector_type(2))) float v2f;
typedef __attribute__((ext_vector_type(4))) float v4f;
typedef __attribute__((ext_vector_type(8))) float v8f;

#define B_   64
#define C_   3
#define H_   512
#define W_   512
#define GH_  128
#define GW_  128
#define KTOP_ 4096

// Order-preserving float -> uint key (descending top-k == largest keys).
__device__ __forceinline__ unsigned fkey(float f) {
    unsigned b = __float_as_uint(f);
    return (b & 0x80000000u) ? ~b : (b | 0x80000000u);
}

// Wait for this wave's outstanding async (global<->LDS) transfers.
__device__ __forceinline__ void wait_async_zero() {
#if __has_builtin(__builtin_amdgcn_s_wait_asynccnt)
    __builtin_amdgcn_s_wait_asynccnt(0);
#else
    asm volatile("s_wait_asynccnt 0x0" ::: "memory");
#endif
}

// ---------------------------------------------------------------------------
// K1: 4x4/stride-4 conv as WMMA. Block = 64 threads (2 waves), handles one
// batch b and 2 gate rows (y = 2*yg + wave). Stages 3ch x 8 input rows x 512
// floats (48 KB) into LDS via async-to-LDS (no VGPR bounce, ASYNCcnt-tracked);
// each global input row is read exactly once across the grid.
//
// WMMA tiling: G_tile[16 pos x 16 cols] += A[16x4] * B[4x16], 12 K-steps
// cover the 48-tap patch. B broadcasts the kernel into all 16 columns, so
// every column of C holds the same 16 results.
//
// Fragment layout (ISA 7.12.2, 32-bit, wave32):
//   A 16x4 : lanes 0-15 -> K={0,1} in v0,v1 ; lanes 16-31 -> K={2,3}
//   B 4x16 : mirrored (lane-half selects upper K pair)        [assumed]
//   C 16x16: VGPR j of lane l = C[j + 8*(l>=16)][l&15]
// ---------------------------------------------------------------------------
__global__ __launch_bounds__(64)
void gate_conv_wmma(const float* __restrict__ in,
                    const float* __restrict__ wk,
                    float* __restrict__ g)
{
    __shared__ float s[C_][8][W_];   // 48 KB
    __shared__ float sw[48];

    const int tid = threadIdx.x;
    const int b   = blockIdx.x >> 6;   // 64 row-groups per batch
    const int yg  = blockIdx.x & 63;   // 2 gate rows per group

    // Stage input rows 8*yg .. 8*yg+7 for all 3 channels.
    // Per-lane GLOBAL_LOAD_ASYNC_TO_LDS_B128: VDST = wave-relative LDS byte
    // offset (low 32 bits of the flat LDS pointer), VADDR = 64-bit global addr.
    for (int i = tid; i < (C_ * 8 * W_) / 4; i += 64) {
        const int c  = i >> 10;         // 1024 vec4 per channel
        const int r  = (i >> 7) & 7;
        const int xq = i & 127;
        const float* src = in +
            (((size_t)b * C_ + c) * H_ + (size_t)(8 * yg + r)) * W_ + 4 * xq;
        const unsigned           lds = (unsigned)(uintptr_t)(&s[c][r][4 * xq]);
        const unsigned long long ga  = (unsigned long long)(uintptr_t)src;
        asm volatile("global_load_async_to_lds_b128 %0, %1, off"
                     :: "v"(lds), "v"(ga) : "memory");
    }
    if (tid < 48) sw[tid] = wk[tid];
    wait_async_zero();      // each wave drains its own ASYNCcnt ...
    __syncthreads();        // ... then all waves rendezvous -> LDS tile valid

    const int wave = tid >> 5;   // local gate row (0..1)
    const int lane = tid & 31;
    const int hi   = lane >> 4;  // K half selector
    const int m    = lane & 15;  // M (output position) within tile

    // Pre-build B fragments: all 16 columns identical = kernel taps.
    float bw0[12], bw1[12];
    #pragma unroll
    for (int t = 0; t < 12; ++t) {
        bw0[t] = sw[4 * t + 2 * hi];
        bw1[t] = sw[4 * t + 2 * hi + 1];
    }

    const int y = yg * 2 + wave;
    float* grow = g + ((size_t)b * GH_ + y) * GW_;

    for (int tx = 0; tx < 8; ++tx) {         // 8 tiles of 16 x-positions
        const int x = 16 * tx + m;           // gate column for this lane's A rows
        v8f acc = {0.f, 0.f, 0.f, 0.f, 0.f, 0.f, 0.f, 0.f};
        #pragma unroll
        for (int t = 0; t < 12; ++t) {
            const int k0 = 4 * t + 2 * hi;   // patch index: k = c*16 + i*4 + j
            const int k1 = k0 + 1;
            v2f a;
            a.x = s[k0 >> 4][4 * wave + ((k0 >> 2) & 3)][4 * x + (k0 & 3)];
            a.y = s[k1 >> 4][4 * wave + ((k1 >> 2) & 3)][4 * x + (k1 & 3)];
            v2f bb;
            bb.x = bw0[t];
            bb.y = bw1[t];
            acc = __builtin_amdgcn_wmma_f32_16x16x4_f32(
                false, a, false, bb, (short)0, acc, false, false);
        }
        // Column 0 lanes (l==0, l==16) hold M=0..7 / M=8..15 in their 8 VGPRs.
        if (m == 0) {
            float* dst = grow + 16 * tx + 8 * hi;
            #pragma unroll
            for (int q = 0; q < 8; ++q) dst[q] = acc[q];
        }
    }
}

// ---------------------------------------------------------------------------
// K2: per-batch threshold. 4-level MSD radix select (256 bins/level) over the
// 16384 gate values -> exact 32-bit key T of the KTOP-th largest element.
// Keep rule downstream: key >= T (exact-K barring full 32-bit float ties).
// ---------------------------------------------------------------------------
__global__ __launch_bounds__(256)
void gate_select(const float* __restrict__ g, unsigned* __restrict__ thr)
{
    __shared__ unsigned hist[256];
    __shared__ unsigned sPrefix, sNeed;

    const int b   = blockIdx.x;
    const int tid = threadIdx.x;
    const float* row = g + (size_t)b * (GH_ * GW_);

    if (tid == 0) { sPrefix = 0u; sNeed = KTOP_; }

    for (int level = 3; level >= 0; --level) {
        hist[tid] = 0u;
        __syncthreads();
        const unsigned prefix = sPrefix;
        const unsigned pmask  = (level == 3) ? 0u
                              : (0xFFFFFFFFu << ((level + 1) * 8));
        const int shift = level * 8;
        for (int i = tid; i < GH_ * GW_; i += 256) {
            const unsigned key = fkey(row[i]);
            if ((key & pmask) == (prefix & pmask))
                atomicAdd(&hist[(key >> shift) & 255u], 1u);
        }
        __syncthreads();
        if (tid == 0) {
            const unsigned need = sNeed;
            unsigned cum = 0;
            int bin = 255;
            for (; bin > 0; --bin) {
                const unsigned c = hist[bin];
                if (cum + c >= need) break;
                cum += c;
            }
            sPrefix = prefix | ((unsigned)bin << shift);
            sNeed   = need - cum;
        }
        __syncthreads();
    }
    if (tid == 0) thr[b] = sPrefix;
}

// ---------------------------------------------------------------------------
// K3: out = in * gate, gate = g[b][h>>2][w>>2] if key >= thr[b] else 0.
// One vec4 per thread (the 4 lanes share one gate value since GKS=4).
// Nontemporal stores: output is write-once, keep it out of L2.
// ---------------------------------------------------------------------------
__global__ __launch_bounds__(256)
void gate_apply(const float* __restrict__ in, const float* __restrict__ g,
                const unsigned* __restrict__ thr, float* __restrict__ out)
{
    const size_t idx = (size_t)blockIdx.x * 256 + threadIdx.x;
    size_t t = idx;
    const int w4 = (int)(t & 127); t >>= 7;   // vec4 column == gate x (w>>2)
    const int h  = (int)(t & 511); t >>= 9;
    t /= 3;                                   // skip channel (gate broadcast)
    const int b  = (int)t;

    const float gv = g[((size_t)b * GH_ + (size_t)(h >> 2)) * GW_ + w4];
    const float gate = (fkey(gv) >= thr[b]) ? gv : 0.0f;

    const v4f v = reinterpret_cast<const v4f*>(in)[idx];
    v4f o = v * gate;
    __builtin_nontemporal_store(o, reinterpret_cast<v4f*>(out) + idx);
}

// ---------------------------------------------------------------------------
extern "C" void kernel_launch(void* const* d_in, const int* in_sizes, int n_in,
                              void* d_out, int out_size, void* d_ws, size_t ws_size,
                              hipStream_t stream)
{
    const float* in = (const float*)d_in[0];   // [64,3,512,512] f32
    const float* wk = (const float*)d_in[1];   // [1,3,4,4] f32
    float* out = (float*)d_out;                // [64,3,512,512] f32

    float*    g   = (float*)d_ws;                                        // 4 MB
    unsigned* thr = (unsigned*)((char*)d_ws +
                                (size_t)B_ * GH_ * GW_ * sizeof(float)); // 256 B

    gate_conv_wmma<<<B_ * 64, 64, 0, stream>>>(in, wk, g);
    gate_select<<<B_, 256, 0, stream>>>(g, thr);

    const size_t nvec = (size_t)B_ * C_ * H_ * W_ / 4;   // 12,582,912
    gate_apply<<<(unsigned)(nvec / 256), 256, 0, stream>>>(in, g, thr, out);
}